// SingleLayerScratchpadPruner_74036646249161
// MI455X (gfx1250) — compile-verified
//
#include <hip/hip_runtime.h>
#include <hip/hip_bf16.h>

typedef __attribute__((ext_vector_type(2))) float v2f;
typedef __attribute__((ext_vector_type(8))) float v8f;

#define B_    4
#define H_    32
#define HKV_  8
#define Q_    16
#define N_    4096
#define NK_   3072
#define D_    128
#define HD_   (H_ * D_)   // 4096

// ---------------------------------------------------------------------------
// Kernel 1: aw[row, j] = attn_w[row, keep_idx[j]] / (sum_j(...) + 1e-6)
// One 256-thread block per (b,h,q) row. 12 elements per thread.
// ---------------------------------------------------------------------------
__global__ void __launch_bounds__(256)
gather_renorm_kernel(const float* __restrict__ attn_w,
                     const int* __restrict__ keep_idx,
                     float* __restrict__ aw) {
    const int row = blockIdx.x;                       // [0, B*H*Q)
    const float* __restrict__ src = attn_w + (size_t)row * N_;
    float* __restrict__ dst = aw + (size_t)row * NK_;

    float vals[12];
    float sum = 0.f;
#pragma unroll
    for (int i = 0; i < 12; ++i) {
        const int j = threadIdx.x + i * 256;
        vals[i] = src[keep_idx[j]];
        sum += vals[i];
    }
    // wave32 reduction
#pragma unroll
    for (int off = 16; off > 0; off >>= 1)
        sum += __shfl_down(sum, off, 32);

    __shared__ float red[8];
    __shared__ float total;
    const int lane = threadIdx.x & 31;
    const int wid  = threadIdx.x >> 5;
    if (lane == 0) red[wid] = sum;
    __syncthreads();
    if (threadIdx.x == 0) {
        float t = 0.f;
#pragma unroll
        for (int w = 0; w < 8; ++w) t += red[w];
        total = 1.0f / (t + 1e-6f);
    }
    __syncthreads();
    const float inv = total;
#pragma unroll
    for (int i = 0; i < 12; ++i)
        dst[threadIdx.x + i * 256] = vals[i] * inv;
}

// ---------------------------------------------------------------------------
// Kernel 2: per (b,h): ctx[16,128] = aw[16,3072] x Vg[3072,128]
// Vg row n = v_cache[b, h/4, keep_idx[n], :] (gather fused into B loads).
// 1 block per (b,h); 8 waves, each owns one 16-wide slice of D.
// fp32 WMMA 16x16x4, K loop of 768 steps.
// ctx stored as [B, Q, H, D] so the reference transpose is free.
// ---------------------------------------------------------------------------
__global__ void __launch_bounds__(256)
ctx_wmma_kernel(const float* __restrict__ aw,
                const float* __restrict__ v_cache,
                const int* __restrict__ keep_idx,
                float* __restrict__ ctx) {
    const int bh    = blockIdx.x;          // 0..127
    const int b     = bh >> 5;
    const int h     = bh & 31;
    const int hkv   = h >> 2;
    const int lane  = threadIdx.x & 31;
    const int n0    = (threadIdx.x >> 5) * 16;   // wave -> D tile
    const int lhalf = lane >> 4;                 // 0: K{0,1}, 1: K{2,3}
    const int l16   = lane & 15;

    const float* __restrict__ A = aw + (size_t)bh * Q_ * NK_;          // [16][3072]
    const float* __restrict__ V = v_cache + ((size_t)b * HKV_ + hkv) * N_ * D_;

    const float* __restrict__ Arow = A + (size_t)l16 * NK_;
    v8f c = {};
#pragma unroll 4
    for (int k0 = 0; k0 < NK_; k0 += 4) {
        const int ka = k0 + lhalf * 2;
        v2f a = *(const v2f*)(Arow + ka);
        const int r0 = keep_idx[ka];
        const int r1 = keep_idx[ka + 1];
        v2f bv;
        bv.x = V[(size_t)r0 * D_ + n0 + l16];
        bv.y = V[(size_t)r1 * D_ + n0 + l16];
        c = __builtin_amdgcn_wmma_f32_16x16x4_f32(false, a, false, bv,
                                                  (short)0, c, false, false);
    }
    // C/D layout: VGPR r, lanes 0-15 -> M=r, lanes 16-31 -> M=r+8
#pragma unroll
    for (int r = 0; r < 8; ++r) {
        const int m = r + lhalf * 8;                 // query index q
        ctx[(((size_t)b * Q_ + m) * H_ + h) * D_ + n0 + l16] = c[r];
    }
}

// ---------------------------------------------------------------------------
// Kernel 3: out[64,4096] = ctx[64,4096] x o_proj_w^T
// out[i,j] = sum_k ctx[i,k] * W[j,k]  ->  B operand loads are contiguous
// float2 reads from W rows. 4 M-tiles x 256 N-tiles; 8 waves/block share
// an M-tile so A rows hit in cache.
// ---------------------------------------------------------------------------
__global__ void __launch_bounds__(256)
out_wmma_kernel(const float* __restrict__ ctx,
                const float* __restrict__ W,
                float* __restrict__ out) {
    const int mt    = blockIdx.x >> 5;                        // 0..3
    const int nt    = ((blockIdx.x & 31) << 3) + (threadIdx.x >> 5); // 0..255
    const int lane  = threadIdx.x & 31;
    const int lhalf = lane >> 4;
    const int l16   = lane & 15;
    const int m0    = mt * 16;
    const int n0    = nt * 16;

    const float* __restrict__ Arow = ctx + (size_t)(m0 + l16) * HD_;
    const float* __restrict__ Brow = W   + (size_t)(n0 + l16) * HD_; // W row j = B col
    v8f c = {};
#pragma unroll 8
    for (int k0 = 0; k0 < HD_; k0 += 4) {
        const int koff = k0 + lhalf * 2;
        v2f a  = *(const v2f*)(Arow + koff);
        v2f bv = *(const v2f*)(Brow + koff);
        c = __builtin_amdgcn_wmma_f32_16x16x4_f32(false, a, false, bv,
                                                  (short)0, c, false, false);
    }
#pragma unroll
    for (int r = 0; r < 8; ++r) {
        const int m = m0 + r + lhalf * 8;
        out[(size_t)m * HD_ + n0 + l16] = c[r];
    }
}

// ---------------------------------------------------------------------------
extern "C" void kernel_launch(void* const* d_in, const int* in_sizes, int n_in,
                              void* d_out, int out_size, void* d_ws, size_t ws_size,
                              hipStream_t stream) {
    const float* attn_w   = (const float*)d_in[0];   // [4,32,16,4096]
    const float* v_cache  = (const float*)d_in[1];   // [4,8,4096,128]
    const float* o_proj_w = (const float*)d_in[2];   // [4096,4096]
    const int*   keep_idx = (const int*)d_in[3];     // [3072]

    float* out = (float*)d_out;                      // [4,16,4096] = 262144 floats
    float* aw  = out + (size_t)B_ * Q_ * HD_;        // [4,32,16,3072] follows
    float* ctx = (float*)d_ws;                       // 262144 floats = 1 MB scratch

    gather_renorm_kernel<<<B_ * H_ * Q_, 256, 0, stream>>>(attn_w, keep_idx, aw);
    ctx_wmma_kernel<<<B_ * H_, 256, 0, stream>>>(aw, v_cache, keep_idx, ctx);
    out_wmma_kernel<<<128, 256, 0, stream>>>(ctx, o_proj_w, out);
}